// KVmemNN_83528523973336
// MI455X (gfx1250) — compile-verified
//
#include <hip/hip_runtime.h>
#include <hip/hip_bf16.h>
#include <math.h>

// ---------------------------------------------------------------------------
// KV-MemNN forward for MI455X (gfx1250, wave32).
// f32 end-to-end; matrix passes use V_WMMA_F32_16X16X4_F32.
// Divisions in the hot path use v_rcp_f32 (__builtin_amdgcn_rcpf) + mul.
// ---------------------------------------------------------------------------

typedef float v2f __attribute__((ext_vector_type(2)));
typedef float v8f __attribute__((ext_vector_type(8)));

#define D_DIM 128
#define EPS 1e-6f

__device__ __forceinline__ float fast_div(float a, float b) {
  return a * __builtin_amdgcn_rcpf(b);   // v_rcp_f32 + v_mul_f32
}

// ---------------------------------------------------------------------------
// EmbeddingBag-sum: out[row] = sum_t table[ids[row*L + t]], optional row norm.
// One wave32 per row; each lane owns a float4 slice of the 128-dim row.
// ---------------------------------------------------------------------------
__global__ void embed_bag_norm(const float* __restrict__ table,
                               const int* __restrict__ ids,
                               float* __restrict__ out,
                               float* __restrict__ norms,
                               int N, int L) {
  const int wave = threadIdx.x >> 5;
  const int lane = threadIdx.x & 31;
  const int row  = blockIdx.x * (blockDim.x >> 5) + wave;
  if (row >= N) return;

  const int* rid = ids + (long)row * L;
  float4 acc = make_float4(0.f, 0.f, 0.f, 0.f);
  for (int t = 0; t < L; ++t) {
    const int id = rid[t];
    if (t + 1 < L) {
      // gfx1250: lowers to global_prefetch_b8 (table is L2-resident, warm L0)
      __builtin_prefetch(table + (long)rid[t + 1] * D_DIM + lane * 4, 0, 1);
    }
    const float4 v = *(const float4*)(table + (long)id * D_DIM + lane * 4);
    acc.x += v.x; acc.y += v.y; acc.z += v.z; acc.w += v.w;
  }
  *(float4*)(out + (long)row * D_DIM + lane * 4) = acc;

  if (norms) {
    float ss = acc.x * acc.x + acc.y * acc.y + acc.z * acc.z + acc.w * acc.w;
    #pragma unroll
    for (int off = 16; off > 0; off >>= 1) ss += __shfl_xor(ss, off, 32);
    if (lane == 0) norms[row] = sqrtf(ss);
  }
}

// ---------------------------------------------------------------------------
// Persona hop + first linear: tiny (N_PERS=20), single block of 128 threads.
// att = softmax(cos(q, pers)); qplus = (q + att@pers) @ W.T
// Also zeroes the 129-float memory-hop accumulator for the next kernel.
// ---------------------------------------------------------------------------
__global__ void persona_hop(const float* __restrict__ q,
                            const float* __restrict__ qnorm_p,
                            const float* __restrict__ enc_pers,
                            const float* __restrict__ pers_norm,
                            const float* __restrict__ W,
                            float* __restrict__ qplus,
                            float* __restrict__ qplus_norm,
                            float* __restrict__ accum,   // [129]
                            int NP) {
  __shared__ float att[32];
  __shared__ float y[D_DIM];
  __shared__ float red[D_DIM];
  const int tid = threadIdx.x;  // 128 threads

  const float qv = q[tid];
  const float qn = qnorm_p[0];

  if (tid < NP) {
    const float* pr = enc_pers + tid * D_DIM;
    float d = 0.f;
    #pragma unroll 8
    for (int k = 0; k < D_DIM; ++k) d += pr[k] * q[k];
    const float den = fmaxf(pers_norm[tid] * qn, EPS);
    att[tid] = __expf(fast_div(d, den));  // cos in [-1,1]: shift-free softmax
  }
  __syncthreads();
  if (tid == 0) {
    float s = 0.f;
    for (int p = 0; p < NP; ++p) s += att[p];
    const float inv = fast_div(1.f, s);
    for (int p = 0; p < NP; ++p) att[p] *= inv;
  }
  __syncthreads();

  float hop = 0.f;
  for (int p = 0; p < NP; ++p) hop += att[p] * enc_pers[p * D_DIM + tid];
  y[tid] = qv + hop;
  __syncthreads();

  const float* wr = W + tid * D_DIM;  // (y @ W.T)[j] = dot(W[j,:], y)
  float o = 0.f;
  #pragma unroll 8
  for (int k = 0; k < D_DIM; ++k) o += wr[k] * y[k];
  qplus[tid] = o;
  red[tid] = o * o;
  __syncthreads();
  for (int s = 64; s > 0; s >>= 1) {
    if (tid < s) red[tid] += red[tid + s];
    __syncthreads();
  }
  if (tid == 0) qplus_norm[0] = sqrtf(red[0]);

  accum[tid] = 0.f;
  if (tid == 0) accum[D_DIM] = 0.f;
}

// ---------------------------------------------------------------------------
// Memory hop (the 65536-row pass). WMMA f32 16x16x4:
//   A = 16 enc_key rows x K=4 chunk, B = q chunk broadcast to all 16 cols.
//   => after 32 chunks, C[r] holds dot(row, q) (lanes 0-15: row base+r,
//      lanes 16-31: row base+8+r).
// Then w = exp(dot / max(|k||q|, eps)); accumulate sum(w) and sum(w * value).
// Block = 256 threads = 8 waves = 128 rows.
// ---------------------------------------------------------------------------
__global__ void mem_hop_wmma(const float* __restrict__ enc_keys,
                             const float* __restrict__ enc_values,
                             const float* __restrict__ key_norm,
                             const float* __restrict__ qp,
                             const float* __restrict__ qp_norm,
                             float* __restrict__ accum,   // [129] global
                             int N) {
  __shared__ float qs[D_DIM];
  __shared__ float lacc[D_DIM + 1];
  const int tid = threadIdx.x;
  if (tid < D_DIM) qs[tid] = qp[tid];
  if (tid <= D_DIM) lacc[tid] = 0.f;
  __syncthreads();

  const int waveId = tid >> 5;
  const int lane   = tid & 31;
  const int half   = lane >> 4;     // 0: K+0..1 / rows r, 1: K+2..3 / rows 8+r
  const int lrow   = lane & 15;
  const int base   = blockIdx.x * 128 + waveId * 16;

  const float* arow = enc_keys + (long)(base + lrow) * D_DIM + half * 2;

  v8f c = {};
  #pragma unroll 8
  for (int k0 = 0; k0 < D_DIM; k0 += 4) {
    const v2f a = *(const v2f*)(arow + k0);                 // global_load_b64
    const v2f b = *(const v2f*)(&qs[k0 + half * 2]);        // ds_load_2addr_b64
    c = __builtin_amdgcn_wmma_f32_16x16x4_f32(false, a, false, b,
                                              (short)0, c, false, false);
  }

  const float qn = qp_norm[0];
  float4 acc = make_float4(0.f, 0.f, 0.f, 0.f);
  float wsum = 0.f;
  const float* vbase = enc_values + lane * 4;   // each lane owns a D-slice

  #pragma unroll
  for (int r = 0; r < 8; ++r) {
    const int myrow = base + half * 8 + r;
    const float sim = fast_div(c[r], fmaxf(key_norm[myrow] * qn, EPS));
    const float w   = __expf(sim);
    const float w0  = __shfl(w, 0, 32);    // weight of row base+r
    const float w1  = __shfl(w, 16, 32);   // weight of row base+8+r
    wsum += w0 + w1;
    const float4 v0 = *(const float4*)(vbase + (long)(base + r) * D_DIM);
    const float4 v1 = *(const float4*)(vbase + (long)(base + 8 + r) * D_DIM);
    acc.x += w0 * v0.x + w1 * v1.x;
    acc.y += w0 * v0.y + w1 * v1.y;
    acc.z += w0 * v0.z + w1 * v1.z;
    acc.w += w0 * v0.w + w1 * v1.w;
  }

  atomicAdd(&lacc[lane * 4 + 0], acc.x);   // ds_add_f32
  atomicAdd(&lacc[lane * 4 + 1], acc.y);
  atomicAdd(&lacc[lane * 4 + 2], acc.z);
  atomicAdd(&lacc[lane * 4 + 3], acc.w);
  if (lane == 0) atomicAdd(&lacc[D_DIM], wsum);
  __syncthreads();

  if (tid <= D_DIM) atomicAdd(&accum[tid], lacc[tid]);
}

// ---------------------------------------------------------------------------
// Finalize hop: q_out = (qplus + sum_wv / wsum) @ W.T, plus its norm.
// ---------------------------------------------------------------------------
__global__ void finalize_hop(const float* __restrict__ qplus,
                             const float* __restrict__ accum,
                             const float* __restrict__ W,
                             float* __restrict__ q_out,
                             float* __restrict__ qnorm_out) {
  __shared__ float y[D_DIM];
  __shared__ float red[D_DIM];
  const int tid = threadIdx.x;  // 128
  const float inv = fast_div(1.f, accum[D_DIM]);
  y[tid] = qplus[tid] + accum[tid] * inv;
  __syncthreads();

  const float* wr = W + tid * D_DIM;
  float o = 0.f;
  #pragma unroll 8
  for (int k = 0; k < D_DIM; ++k) o += wr[k] * y[k];
  q_out[tid] = o;
  red[tid] = o * o;
  __syncthreads();
  for (int s = 64; s > 0; s >>= 1) {
    if (tid < s) red[tid] += red[tid + s];
    __syncthreads();
  }
  if (tid == 0) qnorm_out[0] = sqrtf(red[0]);
}

// ---------------------------------------------------------------------------
// preds = cos(q, enc_cands): same WMMA structure, direct scalar writes.
// OOB rows clamp their loads (EXEC stays all-ones for WMMA) and skip stores.
// ---------------------------------------------------------------------------
__global__ void preds_wmma(const float* __restrict__ enc_cands,
                           const float* __restrict__ cand_norm,
                           const float* __restrict__ q,
                           const float* __restrict__ qnorm,
                           float* __restrict__ out, int N) {
  __shared__ float qs[D_DIM];
  const int tid = threadIdx.x;
  if (tid < D_DIM) qs[tid] = q[tid];
  __syncthreads();

  const int waveId = tid >> 5;
  const int lane   = tid & 31;
  const int half   = lane >> 4;
  const int lrow   = lane & 15;
  const int base   = blockIdx.x * 128 + waveId * 16;

  int arow_idx = base + lrow;
  if (arow_idx >= N) arow_idx = N - 1;
  const float* arow = enc_cands + (long)arow_idx * D_DIM + half * 2;

  v8f c = {};
  #pragma unroll 8
  for (int k0 = 0; k0 < D_DIM; k0 += 4) {
    const v2f a = *(const v2f*)(arow + k0);
    const v2f b = *(const v2f*)(&qs[k0 + half * 2]);
    c = __builtin_amdgcn_wmma_f32_16x16x4_f32(false, a, false, b,
                                              (short)0, c, false, false);
  }

  const float qn = qnorm[0];
  if (lrow == 0) {
    #pragma unroll
    for (int r = 0; r < 8; ++r) {
      const int row = base + half * 8 + r;
      if (row < N) out[row] = fast_div(c[r], fmaxf(cand_norm[row] * qn, EPS));
    }
  }
}

// ---------------------------------------------------------------------------
// Host launcher. Inputs per setup_inputs() order:
// 0 xs(1x32 i32) 1 candidates(10000x20) 2 persona(20x20) 3 keys(65536x20)
// 4 values(65536x20) 5 label(unused) 6 shared_emb(50000x128 f32)
// 7 cand_emb(50000x128 f32) 8 R_W(128x128) 9 R2_W(128x128)
// Output: preds (10000 f32). Workspace need: ~72.5 MB.
// ---------------------------------------------------------------------------
extern "C" void kernel_launch(void* const* d_in, const int* in_sizes, int n_in,
                              void* d_out, int out_size, void* d_ws, size_t ws_size,
                              hipStream_t stream) {
  const int*   xs         = (const int*)d_in[0];
  const int*   candidates = (const int*)d_in[1];
  const int*   persona    = (const int*)d_in[2];
  const int*   keys       = (const int*)d_in[3];
  const int*   values     = (const int*)d_in[4];
  /* d_in[5] = label : unused by the reference computation */
  const float* shared_emb = (const float*)d_in[6];
  const float* cand_emb   = (const float*)d_in[7];
  const float* R_W        = (const float*)d_in[8];
  const float* R2_W       = (const float*)d_in[9];
  float*       out        = (float*)d_out;

  const int NMEM = 65536, NCAND = 10000, NPERS = 20, L = 20, LQ = 32;

  float* ws         = (float*)d_ws;
  float* enc_keys   = ws;                                   // 65536*128
  float* enc_values = enc_keys   + (size_t)NMEM  * D_DIM;   // 65536*128
  float* enc_cands  = enc_values + (size_t)NMEM  * D_DIM;   // 10000*128
  float* enc_pers   = enc_cands  + (size_t)NCAND * D_DIM;   // 20*128
  float* key_norm   = enc_pers   + (size_t)NPERS * D_DIM;   // 65536
  float* cand_norm  = key_norm   + NMEM;                    // 10000
  float* pers_norm  = cand_norm  + NCAND;                   // 20 (+pad)
  float* qbuf       = pers_norm  + 32;                      // 128 (16B aligned)
  float* qnorm      = qbuf       + D_DIM;                   // 1 (+pad)
  float* qplus      = qnorm      + 4;                       // 128
  float* qplus_norm = qplus      + D_DIM;                   // 1 (+pad)
  float* accum      = qplus_norm + 4;                       // 129

  const dim3 b256(256);

  // Phase 1: embedding-bag sums (+ row norms where cosine needs them)
  embed_bag_norm<<<(NMEM  + 7) / 8, b256, 0, stream>>>(shared_emb, keys,       enc_keys,   key_norm,  NMEM,  L);
  embed_bag_norm<<<(NMEM  + 7) / 8, b256, 0, stream>>>(shared_emb, values,     enc_values, nullptr,   NMEM,  L);
  embed_bag_norm<<<(NCAND + 7) / 8, b256, 0, stream>>>(cand_emb,   candidates, enc_cands,  cand_norm, NCAND, L);
  embed_bag_norm<<<(NPERS + 7) / 8, b256, 0, stream>>>(shared_emb, persona,    enc_pers,   pers_norm, NPERS, L);
  embed_bag_norm<<<1,               b256, 0, stream>>>(shared_emb, xs,         qbuf,       qnorm,     1,     LQ);

  // Phase 2: two hops (serial dependency chain)
  const float* Ws[2] = { R_W, R2_W };
  for (int h = 0; h < 2; ++h) {
    persona_hop <<<1,          128,  0, stream>>>(qbuf, qnorm, enc_pers, pers_norm,
                                                  Ws[h], qplus, qplus_norm, accum, NPERS);
    mem_hop_wmma<<<NMEM / 128, b256, 0, stream>>>(enc_keys, enc_values, key_norm,
                                                  qplus, qplus_norm, accum, NMEM);
    finalize_hop<<<1,          128,  0, stream>>>(qplus, accum, Ws[h], qbuf, qnorm);
  }

  // Phase 3: predictions
  preds_wmma<<<(NCAND + 127) / 128, b256, 0, stream>>>(enc_cands, cand_norm,
                                                       qbuf, qnorm, out, NCAND);
}